// PairwiseLoss_33036888441194
// MI455X (gfx1250) — compile-verified
//
#include <hip/hip_runtime.h>

// CDNA5 / gfx1250, wave32. Full-fp32 WMMA path via V_WMMA_F32_16X16X4_F32.
//
//  out = triu((sqrt_guarded(n_i + n_j - 2*G_ij) - y_true)^2),  G = X X^T
//
//  Memory-bound problem (~7MB HBM @ 23.3 TB/s; 0.27 GFLOP). Strategy:
//   1) row_norms_kernel : n_i = ||x_i||^2            (one wave per row)
//   2) pack_kernel      : repack X into fragment-major layout so every
//                         WMMA A/B fragment load is one contiguous 256B
//                         global_load_b64 (vs 16 scattered cache lines)
//   3) pairwise kernel  : 1 wave = 16x64 strip, 4 f32 WMMA accumulators,
//                         A fragment reused 4x; lower-tri blocks skip GEMM.

typedef __attribute__((ext_vector_type(2))) float v2f;
typedef __attribute__((ext_vector_type(8))) float v8f;

#define NN 1024
#define DD 256
#define EPSF 1e-7f

// ---------------------------------------------------------------------------
// Kernel 1: row norms  n[i] = sum_d X[i][d]^2   (one wave per row)
// ---------------------------------------------------------------------------
__global__ __launch_bounds__(256) void row_norms_kernel(
    const float* __restrict__ X, float* __restrict__ norms) {
    const int wave = threadIdx.x >> 5;
    const int lane = threadIdx.x & 31;
    const int row  = blockIdx.x * 8 + wave;          // 128 blocks * 8 waves = 1024 rows
    const float4* xr = (const float4*)(X + row * DD);
    float4 p = xr[lane];
    float4 q = xr[lane + 32];
    float s = p.x*p.x + p.y*p.y + p.z*p.z + p.w*p.w
            + q.x*q.x + q.y*q.y + q.z*q.z + q.w*q.w;
#pragma unroll
    for (int off = 16; off > 0; off >>= 1)
        s += __shfl_xor(s, off, 32);
    if (lane == 0) norms[row] = s;
}

// ---------------------------------------------------------------------------
// Kernel 2: repack X into WMMA fragment-major layout.
//   Fragment (b, k), b = 16-row block (0..63), k = K-step (0..63):
//     lane < 16 : X[b*16 + lane     ][4k + 0 .. 1]
//     lane >= 16: X[b*16 + lane - 16][4k + 2 .. 3]
//   stored contiguously: P[((b*64 + k)*32 + lane)*2 .. +1]   (256B/fragment)
// ---------------------------------------------------------------------------
__global__ __launch_bounds__(256) void pack_kernel(
    const float* __restrict__ X, float* __restrict__ P) {
    const int tid  = blockIdx.x * 256 + threadIdx.x;  // 512 blocks -> 131072 threads
    const int lane = tid & 31;
    const int k    = (tid >> 5) & 63;
    const int b    = tid >> 11;                       // 0..63
    const int lh   = lane >> 4;
    const int ll   = lane & 15;
    v2f v = *(const v2f*)(X + (b * 16 + ll) * DD + 4 * k + 2 * lh);
    *(v2f*)(P + ((size_t)(b * 64 + k) * 32 + lane) * 2) = v;  // coalesced store
}

// ---------------------------------------------------------------------------
// Shared epilogue: distances -> masked squared error
// ---------------------------------------------------------------------------
__device__ __forceinline__ void epilogue_store(
    const v8f acc[4], const float* __restrict__ yt,
    const float* __restrict__ norms, float* __restrict__ out,
    int i0, int j0, int lh, int ll) {
    float ni[8];
#pragma unroll
    for (int v = 0; v < 8; ++v) ni[v] = norms[i0 + v + 8 * lh];
#pragma unroll
    for (int t = 0; t < 4; ++t) {
        const int n = j0 + 16 * t + ll;
        const float nj = norms[n];
#pragma unroll
        for (int v = 0; v < 8; ++v) {
            const int m = i0 + v + 8 * lh;
            float d2 = ni[v] + nj - 2.0f * acc[t][v];
            if (m == n) d2 = 0.0f;               // exact zero on diagonal (ref semantics)
            d2 = fmaxf(d2, 0.0f);                // guard tiny negative cancellation
            // ref: mask=(d2==0); sqrt(d2+mask*EPS)*(1-mask) == d2>0 ? sqrt(d2) : 0
            const float dist = (d2 > 0.0f) ? sqrtf(d2) : 0.0f;
            const float e = dist - __builtin_nontemporal_load(yt + m * NN + n);
            const float r = (n >= m) ? e * e : 0.0f;   // triu
            __builtin_nontemporal_store(r, out + m * NN + n);
        }
    }
}

__device__ __forceinline__ bool zero_block(float* __restrict__ out,
                                           int i0, int j0, int lh, int ll) {
    if (j0 + 63 < i0) {          // strictly lower-triangular block
#pragma unroll
        for (int t = 0; t < 4; ++t) {
            const int n = j0 + 16 * t + ll;
#pragma unroll
            for (int v = 0; v < 8; ++v)
                __builtin_nontemporal_store(0.0f, out + (i0 + v + 8 * lh) * NN + n);
        }
        return true;
    }
    return false;
}

// ---------------------------------------------------------------------------
// Kernel 3a (preferred): GEMM loop over packed fragments — every fragment
// load is one contiguous 256B global_load_b64.
// ---------------------------------------------------------------------------
__global__ __launch_bounds__(32) void pairwise_loss_packed_kernel(
    const float* __restrict__ P, const float* __restrict__ yt,
    const float* __restrict__ norms, float* __restrict__ out) {
    const int ti  = blockIdx.y;        // 16-row tile index  (0..63)
    const int tjb = blockIdx.x;        // 64-col block index (0..15)
    const int i0  = ti * 16;
    const int j0  = tjb * 64;
    const int lane = threadIdx.x;
    const int lh   = lane >> 4;
    const int ll   = lane & 15;

    if (zero_block(out, i0, j0, lh, ll)) return;

    v8f acc[4];
#pragma unroll
    for (int t = 0; t < 4; ++t) acc[t] = (v8f){};

    const float* aBase = P + (size_t)(ti * 64) * 64 + lane * 2;
    const float* bBase[4];
#pragma unroll
    for (int t = 0; t < 4; ++t)
        bBase[t] = P + (size_t)((tjb * 4 + t) * 64) * 64 + lane * 2;

#pragma unroll 4
    for (int k = 0; k < DD / 4; ++k) {
        v2f a  = *(const v2f*)(aBase    + k * 64);
        v2f b0 = *(const v2f*)(bBase[0] + k * 64);
        v2f b1 = *(const v2f*)(bBase[1] + k * 64);
        v2f b2 = *(const v2f*)(bBase[2] + k * 64);
        v2f b3 = *(const v2f*)(bBase[3] + k * 64);
        acc[0] = __builtin_amdgcn_wmma_f32_16x16x4_f32(false, a, false, b0,
                                                       (short)0, acc[0], false, false);
        acc[1] = __builtin_amdgcn_wmma_f32_16x16x4_f32(false, a, false, b1,
                                                       (short)0, acc[1], false, false);
        acc[2] = __builtin_amdgcn_wmma_f32_16x16x4_f32(false, a, false, b2,
                                                       (short)0, acc[2], false, false);
        acc[3] = __builtin_amdgcn_wmma_f32_16x16x4_f32(false, a, false, b3,
                                                       (short)0, acc[3], false, false);
    }

    epilogue_store(acc, yt, norms, out, i0, j0, lh, ll);
}

// ---------------------------------------------------------------------------
// Kernel 3b (fallback, if workspace too small for packed X): direct-address
// fragment loads straight from X.
// ---------------------------------------------------------------------------
__global__ __launch_bounds__(32) void pairwise_loss_direct_kernel(
    const float* __restrict__ X, const float* __restrict__ yt,
    const float* __restrict__ norms, float* __restrict__ out) {
    const int ti  = blockIdx.y;
    const int tjb = blockIdx.x;
    const int i0  = ti * 16;
    const int j0  = tjb * 64;
    const int lane = threadIdx.x;
    const int lh   = lane >> 4;
    const int ll   = lane & 15;

    if (zero_block(out, i0, j0, lh, ll)) return;

    v8f acc[4];
#pragma unroll
    for (int t = 0; t < 4; ++t) acc[t] = (v8f){};

    const float* arow = X + (i0 + ll) * DD + 2 * lh;
    const float* brow[4];
#pragma unroll
    for (int t = 0; t < 4; ++t)
        brow[t] = X + (j0 + 16 * t + ll) * DD + 2 * lh;

#pragma unroll 4
    for (int k = 0; k < DD / 4; ++k) {
        v2f a  = *(const v2f*)(arow    + 4 * k);
        v2f b0 = *(const v2f*)(brow[0] + 4 * k);
        v2f b1 = *(const v2f*)(brow[1] + 4 * k);
        v2f b2 = *(const v2f*)(brow[2] + 4 * k);
        v2f b3 = *(const v2f*)(brow[3] + 4 * k);
        acc[0] = __builtin_amdgcn_wmma_f32_16x16x4_f32(false, a, false, b0,
                                                       (short)0, acc[0], false, false);
        acc[1] = __builtin_amdgcn_wmma_f32_16x16x4_f32(false, a, false, b1,
                                                       (short)0, acc[1], false, false);
        acc[2] = __builtin_amdgcn_wmma_f32_16x16x4_f32(false, a, false, b2,
                                                       (short)0, acc[2], false, false);
        acc[3] = __builtin_amdgcn_wmma_f32_16x16x4_f32(false, a, false, b3,
                                                       (short)0, acc[3], false, false);
    }

    epilogue_store(acc, yt, norms, out, i0, j0, lh, ll);
}

// ---------------------------------------------------------------------------
extern "C" void kernel_launch(void* const* d_in, const int* in_sizes, int n_in,
                              void* d_out, int out_size, void* d_ws, size_t ws_size,
                              hipStream_t stream) {
    (void)in_sizes; (void)n_in; (void)out_size;
    const float* y_true = (const float*)d_in[0];   // (1024,1024)
    const float* y_pred = (const float*)d_in[1];   // (1024, 256)
    float* out   = (float*)d_out;
    float* norms = (float*)d_ws;                   // 1024 floats
    float* packX = norms + NN;                     // NN*DD floats (1 MB)

    row_norms_kernel<<<128, 256, 0, stream>>>(y_pred, norms);

    const size_t need = (size_t)(NN + NN * DD) * sizeof(float);
    dim3 grid(16, 64);                             // 16 col-blocks x 64 row-tiles
    if (ws_size >= need) {
        pack_kernel<<<512, 256, 0, stream>>>(y_pred, packX);
        pairwise_loss_packed_kernel<<<grid, 32, 0, stream>>>(packX, y_true, norms, out);
    } else {
        pairwise_loss_direct_kernel<<<grid, 32, 0, stream>>>(y_pred, y_true, norms, out);
    }
}